// Conv2d_51719996178967
// MI455X (gfx1250) — compile-verified
//
#include <hip/hip_runtime.h>
#include <hip/hip_bf16.h>

typedef __attribute__((ext_vector_type(2))) float v2f;
typedef __attribute__((ext_vector_type(8))) float v8f;

#define C_IN  128
#define H_IN  56
#define W_IN  56
#define N_IMG 32
#define K_OUT 256
#define HW    (H_IN * W_IN)       // 3136
#define M_TOT (N_IMG * HW)        // 100352
#define KD    (C_IN * 9)          // 1152

#define CCHUNK   16                           // channels staged per LDS pass
#define NPAIR    (CCHUNK / 2)                 // 8 channel-pairs
#define WLDS_N   (9 * NPAIR * K_OUT * 2)      // 36864 floats = 144 KB

// Implicit-GEMM 3x3 conv on the dense fp32 matrix pipe (V_WMMA_F32_16X16X4_F32).
// Block = 256 threads = 8 wave32, arranged 2 (M) x 4 (N).
// Each wave computes a 64x64 output tile as a 4x4 grid of 16x16 WMMA tiles.
// Weights are staged per 16-channel chunk into LDS, transposed to
// w[rs][cpair][oc][2] so B fragments are conflict-free contiguous ds_load_b64
// (lanes read consecutive oc) instead of 32-line global gathers.
// Block tile: 128 pixels x 256 out-channels. Grid: 100352/128 = 784.
__global__ __launch_bounds__(256)
void conv3x3_wmma_f32(const float* __restrict__ x,
                      const float* __restrict__ wgt,
                      const float* __restrict__ bias,
                      float* __restrict__ out)
{
    __shared__ float wlds[WLDS_N];            // 147456 bytes

    const int tid    = threadIdx.x;
    const int lane   = tid & 31;
    const int wave   = tid >> 5;     // 0..7
    const int wave_m = wave & 1;     // 2 waves tile M
    const int wave_n = wave >> 1;    // 4 waves tile N

    const int m_base = blockIdx.x * 128 + wave_m * 64;
    const int n_base = wave_n * 64;

    const int lrow  = lane & 15;     // A row / B-D column within a 16-tile
    const int khalf = lane >> 4;     // 0 -> K{0,1}, 1 -> K{2,3}

    // Per-lane A-row (output pixel) decode for the four M subtiles.
    int nimg[4], oh[4], ow[4];
    #pragma unroll
    for (int mt = 0; mt < 4; ++mt) {
        int m    = m_base + mt * 16 + lrow;
        nimg[mt] = m / HW;
        int pix  = m - nimg[mt] * HW;
        oh[mt]   = pix / W_IN;
        ow[mt]   = pix - oh[mt] * W_IN;
    }

    // B columns (output channels) for the four N subtiles.
    int oc[4];
    #pragma unroll
    for (int nt = 0; nt < 4; ++nt) oc[nt] = n_base + nt * 16 + lrow;

    v8f acc[4][4];
    #pragma unroll
    for (int mt = 0; mt < 4; ++mt)
        #pragma unroll
        for (int nt = 0; nt < 4; ++nt) acc[mt][nt] = (v8f){};

    for (int cb0 = 0; cb0 < C_IN; cb0 += CCHUNK) {
        // ---- Stage weights[*, cb0:cb0+16, *, *] into LDS, transposed ----
        __syncthreads();                       // previous chunk fully consumed
        #pragma unroll 1
        for (int idx = tid; idx < K_OUT * (CCHUNK * 9); idx += 256) {
            int ocq = idx / (CCHUNK * 9);      // 0..255
            int t   = idx - ocq * (CCHUNK * 9);// 0..143, contiguous in global
            int cl  = t / 9;                   // local channel 0..15
            int rs  = t - cl * 9;              // tap 0..8
            float w = wgt[ocq * KD + (cb0 + cl) * 9 + rs];   // coalesced read
            int cp  = cl >> 1;
            int j   = cl & 1;
            wlds[(((rs * NPAIR + cp) * K_OUT + ocq) << 1) | j] = w;
        }
        __syncthreads();

        // ---- Consume the chunk: 9 taps x 4 K-steps ----
        for (int rs = 0; rs < 9; ++rs) {
            const int r = rs / 3;
            const int s = rs - r * 3;

            // Padding predicate + clamped base offset, hoisted out of c-loop.
            int  abase[4];
            bool aval[4];
            #pragma unroll
            for (int mt = 0; mt < 4; ++mt) {
                int ih = oh[mt] + r - 1;
                int iw = ow[mt] + s - 1;
                aval[mt]  = ((unsigned)ih < (unsigned)H_IN) & ((unsigned)iw < (unsigned)W_IN);
                int off   = nimg[mt] * (C_IN * HW) + ih * W_IN + iw;
                abase[mt] = aval[mt] ? off : 0;          // clamp to a safe address
            }

            #pragma unroll
            for (int cb = 0; cb < CCHUNK; cb += 4) {
                const int c0 = cb0 + cb + khalf * 2;     // absolute channel
                const int c1 = c0 + 1;

                v2f a[4], b[4];
                #pragma unroll
                for (int mt = 0; mt < 4; ++mt) {
                    float t0 = x[abase[mt] + c0 * HW];
                    float t1 = x[abase[mt] + c1 * HW];
                    a[mt].x = aval[mt] ? t0 : 0.0f;      // v_cndmask, EXEC all-1s
                    a[mt].y = aval[mt] ? t1 : 0.0f;
                }
                const int cpbase = rs * NPAIR + (cb >> 1) + khalf;
                #pragma unroll
                for (int nt = 0; nt < 4; ++nt) {
                    // contiguous across lanes (consecutive oc): ds_load_b64
                    b[nt] = *(const v2f*)&wlds[(cpbase * K_OUT + oc[nt]) << 1];
                }

                #pragma unroll
                for (int mt = 0; mt < 4; ++mt)
                    #pragma unroll
                    for (int nt = 0; nt < 4; ++nt)
                        acc[mt][nt] = __builtin_amdgcn_wmma_f32_16x16x4_f32(
                            false, a[mt], false, b[nt], (short)0, acc[mt][nt],
                            false, false);
            }
        }
    }

    // Epilogue: bias + store.  C/D layout: VGPR v holds row (v + 8*khalf),
    // column = lane&15 within each 16x16 tile.
    #pragma unroll
    for (int nt = 0; nt < 4; ++nt) {
        const float bv = bias[oc[nt]];
        #pragma unroll
        for (int mt = 0; mt < 4; ++mt) {
            #pragma unroll
            for (int v = 0; v < 8; ++v) {
                const int mrow = m_base + mt * 16 + 8 * khalf + v;
                const int ni   = mrow / HW;
                const int pix  = mrow - ni * HW;
                out[ni * (K_OUT * HW) + oc[nt] * HW + pix] = acc[mt][nt][v] + bv;
            }
        }
    }
}

extern "C" void kernel_launch(void* const* d_in, const int* in_sizes, int n_in,
                              void* d_out, int out_size, void* d_ws, size_t ws_size,
                              hipStream_t stream) {
    (void)in_sizes; (void)n_in; (void)out_size; (void)d_ws; (void)ws_size;
    const float* x    = (const float*)d_in[0];   // (32,128,56,56)
    const float* wgt  = (const float*)d_in[1];   // (256,128,3,3)
    const float* bias = (const float*)d_in[2];   // (256,)
    float* out        = (float*)d_out;           // (32,256,56,56)

    dim3 grid(M_TOT / 128, 1);                   // 784 blocks
    dim3 block(256);
    conv3x3_wmma_f32<<<grid, block, 0, stream>>>(x, wgt, bias, out);
}